// RNN_GRU_12489764896810
// MI455X (gfx1250) — compile-verified
//
#include <hip/hip_runtime.h>

#define H 8

// ---- feature probes -------------------------------------------------------
#if defined(__has_builtin)
#  if __has_builtin(__builtin_amdgcn_global_load_async_to_lds_b128)
#    define HAS_ASYNC 1
#  endif
#  if __has_builtin(__builtin_amdgcn_global_store_async_from_lds_b128)
#    define HAS_ASYNC_ST 1
#  endif
#  if __has_builtin(__builtin_amdgcn_s_wait_asynccnt)
#    define HAS_WAITASYNC 1
#  endif
#endif
#ifndef HAS_ASYNC
#  define HAS_ASYNC 0
#endif
#ifndef HAS_ASYNC_ST
#  define HAS_ASYNC_ST 0
#endif
#ifndef HAS_WAITASYNC
#  define HAS_WAITASYNC 0
#endif

#if HAS_WAITASYNC
#  define WAIT_ASYNC(n) __builtin_amdgcn_s_wait_asynccnt(n)
#else
#  define WAIT_ASYNC(n) asm volatile("s_wait_asynccnt %0" ::"i"(n) : "memory")
#endif
#define WAIT_DS0() asm volatile("s_wait_dscnt 0x0" ::: "memory")

// Builtin signatures (confirmed by clang diagnostics): (AS1 v4i*, AS3 v4i*, imm, imm)
typedef int v4i __attribute__((__vector_size__(4 * sizeof(int))));
#define GPTR_V4(p) ((__attribute__((address_space(1))) v4i*)(p))
#define LPTR_V4(p) ((__attribute__((address_space(3))) v4i*)(p))

// ---- fast math on the TRANS pipe -----------------------------------------
__device__ __forceinline__ float sigm(float x) {
  return __builtin_amdgcn_rcpf(1.0f + __builtin_amdgcn_exp2f(-1.4426950408889634f * x));
}
__device__ __forceinline__ float tanh_(float x) {
  return 1.0f - 2.0f * __builtin_amdgcn_rcpf(1.0f + __builtin_amdgcn_exp2f(2.8853900817779268f * x));
}
__device__ __forceinline__ float bcastf(float v, int srclane) {
  return __int_as_float(__builtin_amdgcn_readlane(__float_as_int(v), srclane));
}

// ---- fused 2-layer GRU scan + linear readout, single wave32 ---------------
// Lane j (mod 8) owns hidden unit j of each layer; weights live in VGPRs.
// Loop body is software-pipelined: it computes layer1(t) and layer0(t+1),
// which are independent chains, so the in-order SIMD overlaps them.
__global__ __launch_bounds__(32, 1)
void gru2_scan(const float* __restrict__ x, const float* __restrict__ h0,
               const float* __restrict__ wih0, const float* __restrict__ whh0,
               const float* __restrict__ bih0, const float* __restrict__ bhh0,
               const float* __restrict__ wih1, const float* __restrict__ whh1,
               const float* __restrict__ bih1, const float* __restrict__ bhh1,
               const float* __restrict__ wlin, const float* __restrict__ blin,
               float* __restrict__ out, int T) {
  const int lane = threadIdx.x;
  const int j = lane & 7;

  float whh0r[H], whh0z[H], whh0n[H];
  float wih1r[H], wih1z[H], wih1n[H];
  float whh1r[H], whh1z[H], whh1n[H];
  float wl[H], h0b[H], h1b[H];
#pragma unroll
  for (int k = 0; k < H; ++k) {
    whh0r[k] = whh0[(j) * H + k];
    whh0z[k] = whh0[(H + j) * H + k];
    whh0n[k] = whh0[(2 * H + j) * H + k];
    wih1r[k] = wih1[(j) * H + k];
    wih1z[k] = wih1[(H + j) * H + k];
    wih1n[k] = wih1[(2 * H + j) * H + k];
    whh1r[k] = whh1[(j) * H + k];
    whh1z[k] = whh1[(H + j) * H + k];
    whh1n[k] = whh1[(2 * H + j) * H + k];
    wl[k]  = wlin[k];
    h0b[k] = h0[k];
    h1b[k] = h0[H + k];
  }
  const float wxr = wih0[j], wxz = wih0[H + j], wxn = wih0[2 * H + j];
  const float c0r = bih0[j] + bhh0[j];
  const float c0z = bih0[H + j] + bhh0[H + j];
  const float bx0n = bih0[2 * H + j], bh0n = bhh0[2 * H + j];
  const float c1r = bih1[j] + bhh1[j];
  const float c1z = bih1[H + j] + bhh1[H + j];
  const float bx1n = bih1[2 * H + j], bh1n = bhh1[2 * H + j];
  const float bl = blin[0];
  float hown0 = h0[j], hown1 = h0[H + j];

  // Layer-0-only step (pipeline prologue): h0[t] -> h0[t+1] wrt input xt.
  auto l0_step = [&](float xt) {
    float ar = fmaf(wxr, xt, c0r), az = fmaf(wxz, xt, c0z);
    float axn = fmaf(wxn, xt, bx0n), ahn = bh0n;
#pragma unroll
    for (int k = 0; k < H; ++k) {
      ar  = fmaf(whh0r[k], h0b[k], ar);
      az  = fmaf(whh0z[k], h0b[k], az);
      ahn = fmaf(whh0n[k], h0b[k], ahn);
    }
    float r = sigm(ar), z = sigm(az);
    float n = tanh_(fmaf(r, ahn, axn));
    hown0 = fmaf(z, hown0 - n, n);
#pragma unroll
    for (int k = 0; k < H; ++k) h0b[k] = bcastf(hown0, k);
  };

  // Pipelined body: consumes h0b==y0[t], h1b==h1[t-1]; produces out[t],
  // h1[t], and h0[t+1] (needs x[t+1]=xnext). Split accumulators shorten the
  // dependence chains; L0(t+1) and L1(t) interleave in one basic block.
  auto body = [&](float xnext) -> float {
    float br_a = c1r,  br_b = 0.f, bz_a = c1z,  bz_b = 0.f;
    float bxn_a = bx1n, bxn_b = 0.f, bhn_a = bh1n, bhn_b = 0.f;
    float ar_a = fmaf(wxr, xnext, c0r), ar_b = 0.f;
    float az_a = fmaf(wxz, xnext, c0z), az_b = 0.f;
    float axn  = fmaf(wxn, xnext, bx0n);
    float ahn_a = bh0n, ahn_b = 0.f;
#pragma unroll
    for (int k = 0; k < H; k += 2) {
      ar_a  = fmaf(whh0r[k],     h0b[k],     ar_a);
      ar_b  = fmaf(whh0r[k + 1], h0b[k + 1], ar_b);
      az_a  = fmaf(whh0z[k],     h0b[k],     az_a);
      az_b  = fmaf(whh0z[k + 1], h0b[k + 1], az_b);
      ahn_a = fmaf(whh0n[k],     h0b[k],     ahn_a);
      ahn_b = fmaf(whh0n[k + 1], h0b[k + 1], ahn_b);
      br_a  = fmaf(wih1r[k],     h0b[k],     br_a);
      br_b  = fmaf(wih1r[k + 1], h0b[k + 1], br_b);
      br_a  = fmaf(whh1r[k],     h1b[k],     br_a);
      br_b  = fmaf(whh1r[k + 1], h1b[k + 1], br_b);
      bz_a  = fmaf(wih1z[k],     h0b[k],     bz_a);
      bz_b  = fmaf(wih1z[k + 1], h0b[k + 1], bz_b);
      bz_a  = fmaf(whh1z[k],     h1b[k],     bz_a);
      bz_b  = fmaf(whh1z[k + 1], h1b[k + 1], bz_b);
      bxn_a = fmaf(wih1n[k],     h0b[k],     bxn_a);
      bxn_b = fmaf(wih1n[k + 1], h0b[k + 1], bxn_b);
      bhn_a = fmaf(whh1n[k],     h1b[k],     bhn_a);
      bhn_b = fmaf(whh1n[k + 1], h1b[k + 1], bhn_b);
    }
    float r1 = sigm(br_a + br_b), z1 = sigm(bz_a + bz_b);
    float n1 = tanh_(fmaf(r1, bhn_a + bhn_b, bxn_a + bxn_b));
    hown1 = fmaf(z1, hown1 - n1, n1);
    float r = sigm(ar_a + ar_b), z = sigm(az_a + az_b);
    float n = tanh_(fmaf(r, ahn_a + ahn_b, axn));
    hown0 = fmaf(z, hown0 - n, n);
#pragma unroll
    for (int k = 0; k < H; ++k) h0b[k] = bcastf(hown0, k);
#pragma unroll
    for (int k = 0; k < H; ++k) h1b[k] = bcastf(hown1, k);
    float o = bl;
#pragma unroll
    for (int k = 0; k < H; ++k) o = fmaf(wl[k], h1b[k], o);
    return o;  // uniform across lanes
  };

  const int CH = 128;  // 512 B: one async b128 op per chunk per direction
#if HAS_ASYNC
  __shared__ float xbuf[3][CH];  // triple buffer: body reads x across chunk edge
  __shared__ float obuf[2][CH];
  const int nch = T / CH;
  if (nch > 0)
    __builtin_amdgcn_global_load_async_to_lds_b128(
        GPTR_V4(x + lane * 4), LPTR_V4(&xbuf[0][lane * 4]), 0, 0);
  if (nch > 1)
    __builtin_amdgcn_global_load_async_to_lds_b128(
        GPTR_V4(x + CH + lane * 4), LPTR_V4(&xbuf[1][lane * 4]), 0, 0);
  if (nch > 1) { WAIT_ASYNC(1); } else { WAIT_ASYNC(0); }
  if (nch > 0) l0_step(xbuf[0][0]);
  else if (T > 0) l0_step(x[0]);

  for (int c = 0; c < nch; ++c) {
    if (c + 2 < nch) {
      __builtin_amdgcn_global_load_async_to_lds_b128(
          GPTR_V4(x + (c + 2) * CH + lane * 4),
          LPTR_V4(&xbuf[(c + 2) % 3][lane * 4]), 0, 0);
      WAIT_ASYNC(1);  // loads retire in order: chunks c and c+1 resident
    } else {
      WAIT_ASYNC(0);
    }
    const float* xb  = xbuf[c % 3];
    const float* xbn = xbuf[(c + 1) % 3];
#if HAS_ASYNC_ST
    float* ob = obuf[c & 1];
#endif
    for (int tt = 0; tt < CH - 1; ++tt) {
      float o = body(xb[tt + 1]);
#if HAS_ASYNC_ST
      if (lane == 0) ob[tt] = o;
#else
      if (lane == 0) out[c * CH + tt] = o;
#endif
    }
    {  // last step of chunk: x[t+1] lives in the next chunk (or the tail)
      float xnext = (c + 1 < nch) ? xbn[0]
                    : ((nch * CH < T) ? x[nch * CH] : 0.f);
      float o = body(xnext);
#if HAS_ASYNC_ST
      if (lane == 0) ob[CH - 1] = o;
#else
      if (lane == 0) out[c * CH + CH - 1] = o;
#endif
    }
#if HAS_ASYNC_ST
    WAIT_DS0();  // LDS stores visible before the async engine reads them
    __builtin_amdgcn_global_store_async_from_lds_b128(
        GPTR_V4(out + c * CH + lane * 4), LPTR_V4(&ob[lane * 4]), 0, 0);
    // obuf[c&1] is rewritten at chunk c+2; the WAIT_ASYNC above (<=1 or 0)
    // plus in-order store retirement guarantees this flush has drained by then.
#endif
  }
  for (int t = nch * CH; t < T; ++t) {  // tail (empty for T = 131072)
    float o = body((t + 1 < T) ? x[t + 1] : 0.f);
    if (lane == 0) out[t] = o;
  }
  // s_endpgm performs an implicit wait-idle, draining the last async store.
#else
  if (T > 0) l0_step(x[0]);
  for (int t = 0; t < T; ++t) {
    __builtin_prefetch(x + t + 512, 0, 3);
    float o = body((t + 1 < T) ? x[t + 1] : 0.f);
    if (lane == 0) out[t] = o;
  }
#endif
}

extern "C" void kernel_launch(void* const* d_in, const int* in_sizes, int n_in,
                              void* d_out, int out_size, void* d_ws, size_t ws_size,
                              hipStream_t stream) {
  const float* x    = (const float*)d_in[0];
  const float* h0   = (const float*)d_in[1];
  const float* wih0 = (const float*)d_in[2];
  const float* whh0 = (const float*)d_in[3];
  const float* bih0 = (const float*)d_in[4];
  const float* bhh0 = (const float*)d_in[5];
  const float* wih1 = (const float*)d_in[6];
  const float* whh1 = (const float*)d_in[7];
  const float* bih1 = (const float*)d_in[8];
  const float* bhh1 = (const float*)d_in[9];
  const float* wlin = (const float*)d_in[10];
  const float* blin = (const float*)d_in[11];
  float* out = (float*)d_out;
  const int T = in_sizes[0];
  gru2_scan<<<1, 32, 0, stream>>>(x, h0, wih0, whh0, bih0, bhh0,
                                  wih1, whh1, bih1, bhh1, wlin, blin, out, T);
  (void)n_in; (void)out_size; (void)d_ws; (void)ws_size;
}